// HierAdaptiveEmbeddingI2T_88029649698903
// MI455X (gfx1250) — compile-verified
//
#include <hip/hip_runtime.h>
#include <math.h>

typedef float v2f __attribute__((ext_vector_type(2)));
typedef float v8f __attribute__((ext_vector_type(8)));
typedef unsigned int u32x4 __attribute__((ext_vector_type(4)));
typedef int i32x4 __attribute__((ext_vector_type(4)));
typedef int i32x8 __attribute__((ext_vector_type(8)));

#define LEAKY(x) ((x) >= 0.f ? (x) : 0.1f * (x))

#define HAVE_TDM __has_builtin(__builtin_amdgcn_tensor_load_to_lds)

__device__ __forceinline__ unsigned fenc(float f) {
  unsigned u = __float_as_uint(f);
  return (u & 0x80000000u) ? ~u : (u | 0x80000000u);
}
__device__ __forceinline__ float fdec(unsigned k) {
  unsigned u = (k & 0x80000000u) ? (k & 0x7fffffffu) : ~k;
  return __uint_as_float(u);
}

#if HAVE_TDM
// ---------------------------------------------------------------------------
// Tensor Data Mover: DMA a 2D tile (tile_d1 rows x tile_d0 dwords, row stride
// stride_d0 dwords) from global into LDS at lds_byte_off.  With pad_en=1,
// pad_interval_code=4 (32 dwords) and pad_amount_code=1 (2 dwords) the LDS
// rows land with stride 34 dwords -> bank-conflict-free fragment loads.
// D# layout per CDNA5 ISA 8.3/8.4; groups 2/3 zero (2-D tensor).
// This toolchain uses the 6-arg builtin: (g0, g1, g2, g3, g4, cpol).
// ---------------------------------------------------------------------------
__device__ __forceinline__ void tdm_load_2d(
    unsigned lds_byte_off, const void* gptr,
    unsigned tile_d0, unsigned tile_d1,
    unsigned stride_d0,
    unsigned pad_en, unsigned pad_interval_code, unsigned pad_amount_code) {
  unsigned long long ga = (unsigned long long)gptr;
  u32x4 g0;
  g0[0] = 1u;                                    // count=1, user descriptor
  g0[1] = lds_byte_off;                          // lds_addr (bytes)
  g0[2] = (unsigned)(ga & 0xffffffffu);          // global_addr[31:0]
  g0[3] = (unsigned)((ga >> 32) & 0x1ffffffu)    // global_addr[56:32]
          | (2u << 30);                          // type = 2 ("image")
  i32x8 g1;
  g1[0] = (int)((2u << 16)                       // data_size = 4 bytes
                | (pad_en << 20)
                | (pad_interval_code << 22)
                | (pad_amount_code << 25));
  g1[1] = (int)((tile_d0 & 0xffffu) << 16);      // tensor_dim0 = tile_d0
  g1[2] = (int)((tile_d0 >> 16) | ((tile_d1 & 0xffffu) << 16));  // tensor_dim1
  g1[3] = (int)((tile_d1 >> 16) | (tile_d0 << 16));              // tile_dim0
  g1[4] = (int)tile_d1;                          // tile_dim1 (tile_dim2 = 0)
  g1[5] = (int)stride_d0;                        // tensor_dim0_stride[31:0]
  g1[6] = 0;
  g1[7] = 0;
  i32x4 z4;
  z4[0] = 0; z4[1] = 0; z4[2] = 0; z4[3] = 0;
  i32x8 z8;
  for (int k = 0; k < 8; ++k) z8[k] = 0;
  __builtin_amdgcn_tensor_load_to_lds(g0, g1, z4, z4, z8, 0);
}
#endif

// ---------------------------------------------------------------------------
// C[M,N] = act(A[M,K] @ B[N,K]^T + bias[N]).  M%128==0, N%64==0, K%32==0.
// Block = 4 waves; block tile 128x64; wave tile 32x64; f32 WMMA 16x16x4.
// K-chunks of 32 staged to LDS by the TDM, double buffered on TENSORcnt.
// ---------------------------------------------------------------------------
__global__ __launch_bounds__(128) void gemm_nt_wmma(
    const float* __restrict__ A, const float* __restrict__ B,
    const float* __restrict__ bias, float* __restrict__ C,
    int M, int N, int K, int leaky) {
  __shared__ float As[2][128 * 34];
  __shared__ float Bs[2][64 * 34];
  const int tid = threadIdx.x;
  const int lane = tid & 31;
  const int wv = tid >> 5;
  const int m0 = blockIdx.y * 128;
  const int n0 = blockIdx.x * 64;
  const int half = lane >> 4;
  const int mr = lane & 15;
  const int nchunk = K >> 5;

  v8f zz;
  for (int v = 0; v < 8; ++v) zz[v] = 0.f;
  v8f acc[2][4];
  for (int p = 0; p < 2; ++p)
    for (int u = 0; u < 4; ++u) acc[p][u] = zz;

#if HAVE_TDM
  const unsigned asoff = (unsigned)(unsigned long long)(const void*)&As[0][0];
  const unsigned bsoff = (unsigned)(unsigned long long)(const void*)&Bs[0][0];
  if (wv == 0) {
    tdm_load_2d(asoff, A + (size_t)m0 * K, 32u, 128u, (unsigned)K, 1u, 4u, 1u);
    tdm_load_2d(bsoff, B + (size_t)n0 * K, 32u, 64u, (unsigned)K, 1u, 4u, 1u);
  }
#endif

  for (int ch = 0; ch < nchunk; ++ch) {
    const int buf = ch & 1;
#if HAVE_TDM
    if (wv == 0) {
      if (ch + 1 < nchunk) {
        const int k1 = (ch + 1) << 5;
        tdm_load_2d(asoff + (buf ^ 1) * (unsigned)(128 * 34 * 4),
                    A + (size_t)m0 * K + k1, 32u, 128u, (unsigned)K, 1u, 4u, 1u);
        tdm_load_2d(bsoff + (buf ^ 1) * (unsigned)(64 * 34 * 4),
                    B + (size_t)n0 * K + k1, 32u, 64u, (unsigned)K, 1u, 4u, 1u);
        __builtin_amdgcn_s_wait_tensorcnt((short)2);  // current chunk landed
      } else {
        __builtin_amdgcn_s_wait_tensorcnt((short)0);
      }
    }
    __syncthreads();
#else
    {  // fallback: VGPR staging
      const int k0 = ch << 5;
#pragma unroll
      for (int it = 0; it < 8; ++it) {
        int idx = tid * 4 + it * 512;
        int r = idx >> 5, c = idx & 31;
        float4 v4 = *(const float4*)(A + (size_t)(m0 + r) * K + k0 + c);
        float* dst = &As[buf][r * 34 + c];
        dst[0] = v4.x; dst[1] = v4.y; dst[2] = v4.z; dst[3] = v4.w;
      }
#pragma unroll
      for (int it = 0; it < 4; ++it) {
        int idx = tid * 4 + it * 512;
        int r = idx >> 5, c = idx & 31;
        float4 v4 = *(const float4*)(B + (size_t)(n0 + r) * K + k0 + c);
        float* dst = &Bs[buf][r * 34 + c];
        dst[0] = v4.x; dst[1] = v4.y; dst[2] = v4.z; dst[3] = v4.w;
      }
      __syncthreads();
    }
#endif
#pragma unroll
    for (int kk = 0; kk < 32; kk += 4) {
      const int kc = kk + 2 * half;
      v2f a0 = *(const v2f*)&As[buf][(32 * wv + mr) * 34 + kc];
      v2f a1 = *(const v2f*)&As[buf][(32 * wv + 16 + mr) * 34 + kc];
#pragma unroll
      for (int u = 0; u < 4; ++u) {
        v2f bf = *(const v2f*)&Bs[buf][(16 * u + mr) * 34 + kc];
        acc[0][u] = __builtin_amdgcn_wmma_f32_16x16x4_f32(
            false, a0, false, bf, (short)0, acc[0][u], false, false);
        acc[1][u] = __builtin_amdgcn_wmma_f32_16x16x4_f32(
            false, a1, false, bf, (short)0, acc[1][u], false, false);
      }
    }
    __syncthreads();
  }

#pragma unroll
  for (int p = 0; p < 2; ++p)
#pragma unroll
    for (int u = 0; u < 4; ++u) {
      int col = n0 + 16 * u + mr;
      float bv = bias ? bias[col] : 0.f;
#pragma unroll
      for (int v = 0; v < 8; ++v) {
        int row = m0 + 32 * wv + 16 * p + 8 * half + v;
        float x = acc[p][u][v] + bv;
        if (leaky) x = LEAKY(x);
        C[(size_t)row * N + col] = x;
      }
    }
}

// ---------------------------------------------------------------------------
// Image attention: energy->softmax->column sums; vec[b,o]=mean_t(g*out + X)
// collapses to (g * sum_j w[j] V[j,o] + sum_i X[i,o]) / 36.
// ---------------------------------------------------------------------------
__global__ __launch_bounds__(128) void img_attn_vec(
    const float* __restrict__ Q, const float* __restrict__ Km,
    const float* __restrict__ V, const float* __restrict__ X,
    const float* __restrict__ gamma, float* __restrict__ vec) {
  const int b = blockIdx.x;
  const int tid = threadIdx.x;
  __shared__ float e[36][36];
  __shared__ float wsum[36];

  for (int idx = tid; idx < 36 * 36; idx += 128) {
    int i = idx / 36, j = idx % 36;
    const float* q = Q + (size_t)(b * 36 + i) * 128;
    const float* k = Km + (size_t)(b * 36 + j) * 128;
    float s = 0.f;
    for (int a = 0; a < 128; ++a) s += q[a] * k[a];
    e[i][j] = s;
  }
  __syncthreads();
  if (tid < 36) {
    float mx = -__builtin_inff();
    for (int j = 0; j < 36; ++j) mx = fmaxf(mx, e[tid][j]);
    float s = 0.f;
    for (int j = 0; j < 36; ++j) { float ex = __expf(e[tid][j] - mx); e[tid][j] = ex; s += ex; }
    float inv = 1.f / s;
    for (int j = 0; j < 36; ++j) e[tid][j] *= inv;
  }
  __syncthreads();
  if (tid < 36) {
    float s = 0.f;
    for (int i = 0; i < 36; ++i) s += e[i][tid];
    wsum[tid] = s;
  }
  __syncthreads();
  const float g = gamma[0];
  for (int o = tid; o < 1024; o += 128) {
    float s = 0.f, xs = 0.f;
    for (int j = 0; j < 36; ++j) s += wsum[j] * V[(size_t)(b * 36 + j) * 1024 + o];
    for (int i = 0; i < 36; ++i) xs += X[(size_t)(b * 36 + i) * 1024 + o];
    vec[b * 1024 + o] = (g * s + xs) * (1.f / 36.f);
  }
}

// cap_embed (64,50,300) -> capT (64,300,40)
__global__ void build_capT(const float* __restrict__ cap_embed, float* __restrict__ capT) {
  int idx = blockIdx.x * blockDim.x + threadIdx.x;
  if (idx >= 64 * 300 * 40) return;
  int t = idx % 40;
  int c = (idx / 40) % 300;
  int b = idx / (40 * 300);
  capT[idx] = cap_embed[(size_t)b * 50 * 300 + (size_t)t * 300 + c];
}

// out[b,o,t] = sum_c W[o,c] x[b,c,t] + bias[o]   (x: (B,Cin,T))
__global__ void conv1x1_ct(const float* __restrict__ x, const float* __restrict__ W,
                           const float* __restrict__ bias, float* __restrict__ out,
                           int B, int Cin, int T, int O) {
  int idx = blockIdx.x * blockDim.x + threadIdx.x;
  int total = B * O * T;
  if (idx >= total) return;
  int t = idx % T;
  int o = (idx / T) % O;
  int b = idx / (T * O);
  const float* xc = x + (size_t)b * Cin * T + t;
  const float* wr = W + (size_t)o * Cin;
  float s = bias[o];
  for (int c = 0; c < Cin; ++c) s += wr[c] * xc[(size_t)c * T];
  out[(size_t)b * O * T + (size_t)o * T + t] = s;
}

// word attention: energy(40x40, K=75) -> softmax -> cap2 = g*out + capT
__global__ __launch_bounds__(128) void word_attn(
    const float* __restrict__ qw, const float* __restrict__ kw,
    const float* __restrict__ vw, const float* __restrict__ capT,
    const float* __restrict__ gamma, float* __restrict__ cap2) {
  const int b = blockIdx.x;
  const int tid = threadIdx.x;
  __shared__ float e[40][40];
  for (int idx = tid; idx < 40 * 40; idx += 128) {
    int i = idx / 40, j = idx % 40;
    float s = 0.f;
    const float* qp = qw + (size_t)b * 75 * 40;
    const float* kp = kw + (size_t)b * 75 * 40;
    for (int a = 0; a < 75; ++a) s += qp[a * 40 + i] * kp[a * 40 + j];
    e[i][j] = s;
  }
  __syncthreads();
  if (tid < 40) {
    float mx = -__builtin_inff();
    for (int j = 0; j < 40; ++j) mx = fmaxf(mx, e[tid][j]);
    float s = 0.f;
    for (int j = 0; j < 40; ++j) { float ex = __expf(e[tid][j] - mx); e[tid][j] = ex; s += ex; }
    float inv = 1.f / s;
    for (int j = 0; j < 40; ++j) e[tid][j] *= inv;
  }
  __syncthreads();
  const float g = gamma[0];
  for (int idx = tid; idx < 300 * 40; idx += 128) {
    int c = idx / 40, t = idx % 40;
    const float* vp = vw + (size_t)b * 300 * 40 + (size_t)c * 40;
    float s = 0.f;
    for (int j = 0; j < 40; ++j) s += vp[j] * e[t][j];
    size_t off = (size_t)b * 300 * 40 + (size_t)c * 40 + t;
    cap2[off] = g * s + capT[off];
  }
}

// per-channel BN stats over (b,t): 2560 elements per channel
__global__ __launch_bounds__(256) void bn_stats(const float* __restrict__ cap2,
                                                float* __restrict__ chm,
                                                float* __restrict__ chinv) {
  const int c = blockIdx.x;
  const int tid = threadIdx.x;
  __shared__ float rs[256], rq[256];
  float s = 0.f, q = 0.f;
  for (int idx = tid; idx < 2560; idx += 256) {
    int b = idx / 40, t = idx % 40;
    float v = cap2[(size_t)b * 300 * 40 + (size_t)c * 40 + t];
    s += v; q += v * v;
  }
  rs[tid] = s; rq[tid] = q;
  __syncthreads();
  for (int st = 128; st > 0; st >>= 1) {
    if (tid < st) { rs[tid] += rs[tid + st]; rq[tid] += rq[tid + st]; }
    __syncthreads();
  }
  if (tid == 0) {
    float m = rs[0] * (1.f / 2560.f);
    float var = rq[0] * (1.f / 2560.f) - m * m;
    chm[c] = m;
    chinv[c] = rsqrtf(var + 1e-5f);
  }
}

// An[(b*40+t)*300 + c] = (cap2[b,c,t] - m[c]) * inv[c]
__global__ void build_An(const float* __restrict__ cap2, const float* __restrict__ chm,
                         const float* __restrict__ chinv, float* __restrict__ An) {
  int idx = blockIdx.x * blockDim.x + threadIdx.x;
  if (idx >= 2560 * 300) return;
  int c = idx % 300;
  int n = idx / 300;
  int b = n / 40, t = n % 40;
  An[idx] = (cap2[(size_t)b * 300 * 40 + (size_t)c * 40 + t] - chm[c]) * chinv[c];
}

// out[i,:O] = relu(vec[i] @ W1^T + b1) @ W2^T + b2   (hidden = 128)
__global__ __launch_bounds__(128) void mlp2(const float* __restrict__ vec,
                                            const float* __restrict__ W1, const float* __restrict__ b1,
                                            const float* __restrict__ W2, const float* __restrict__ b2,
                                            float* __restrict__ out, int O) {
  const int i = blockIdx.x;
  const int tid = threadIdx.x;
  __shared__ float h[128];
  const float* v = vec + (size_t)i * 1024;
  {
    float s = b1[tid];
    const float* w = W1 + (size_t)tid * 1024;
    for (int o = 0; o < 1024; ++o) s += w[o] * v[o];
    h[tid] = s > 0.f ? s : 0.f;
  }
  __syncthreads();
  for (int o = tid; o < O; o += 128) {
    float s = b2[o];
    const float* w = W2 + (size_t)o * 128;
    for (int a = 0; a < 128; ++a) s += w[a] * h[a];
    out[(size_t)i * O + o] = s;
  }
}

// bprime[i,o] = sum_c Wwp[o,c]*b1[i,c] + bwp[o]
__global__ __launch_bounds__(128) void bprime_k(const float* __restrict__ Wwp,
                                                const float* __restrict__ bwp,
                                                const float* __restrict__ b1arr,
                                                float* __restrict__ bprime) {
  const int i = blockIdx.y;
  const int o = blockIdx.x * 128 + threadIdx.x;
  __shared__ float br[300];
  for (int c = threadIdx.x; c < 300; c += 128) br[c] = b1arr[(size_t)i * 300 + c];
  __syncthreads();
  float s = bwp[o];
  const float* w = Wwp + (size_t)o * 300;
  for (int c = 0; c < 300; ++c) s += w[c] * br[c];
  bprime[(size_t)i * 1024 + o] = s;
}

// iv = l2norm(vec) per image
__global__ __launch_bounds__(256) void iv_norm(const float* __restrict__ vec, float* __restrict__ iv) {
  const int i = blockIdx.x;
  const int tid = threadIdx.x;
  __shared__ float red[256];
  float s = 0.f;
  for (int o = tid; o < 1024; o += 256) { float v = vec[(size_t)i * 1024 + o]; s += v * v; }
  red[tid] = s;
  __syncthreads();
  for (int st = 128; st > 0; st >>= 1) {
    if (tid < st) red[tid] += red[tid + st];
    __syncthreads();
  }
  const float inv = rsqrtf(red[0]);
  __syncthreads();
  for (int o = tid; o < 1024; o += 256) iv[(size_t)i * 1024 + o] = vec[(size_t)i * 1024 + o] * inv;
}

__global__ void zero_k(float* p, int n) {
  int idx = blockIdx.x * blockDim.x + threadIdx.x;
  if (idx < n) p[idx] = 0.f;
}

// ---------------------------------------------------------------------------
// Fused per-image GEMM + conditional-BN + max_t + l2norm-dot.
// grid = (16 o-blocks, 64 images), block = 128 (4 waves).
// Each block owns (i, o in [o0,o0+64)) and sweeps all n = b*40+t (2560 rows).
// ---------------------------------------------------------------------------
__global__ __launch_bounds__(128) void cbn_gemm_reduce(
    const float* __restrict__ An, const float* __restrict__ Wwp,
    const float* __restrict__ g1, const float* __restrict__ g2arr,
    const float* __restrict__ b2arr, const float* __restrict__ bprime,
    const float* __restrict__ iv, float* __restrict__ dotp, float* __restrict__ ssq) {
  const int i = blockIdx.y;
  const int o0 = blockIdx.x * 64;
  const int tid = threadIdx.x;
  const int lane = tid & 31;
  const int w = tid >> 5;
  const int half = lane >> 4;
  const int mr = lane & 15;

  __shared__ float gl[304];          // 1 + g1[i, c]
  __shared__ unsigned mxk[64 * 64];  // encoded max over t per (b, o_local)
  __shared__ unsigned mnk[64 * 64];  // encoded min
  __shared__ float osum[64], osq[64];

  for (int c = tid; c < 300; c += 128) gl[c] = 1.0f + g1[(size_t)i * 300 + c];
  for (int x = tid; x < 64 * 64; x += 128) { mxk[x] = 0u; mnk[x] = 0xFFFFFFFFu; }
  if (tid < 64) { osum[tid] = 0.f; osq[tid] = 0.f; }
  __syncthreads();

  float bp[4];
#pragma unroll
  for (int u = 0; u < 4; ++u) bp[u] = bprime[(size_t)i * 1024 + o0 + 16 * u + mr];
  float psum[4] = {0.f, 0.f, 0.f, 0.f}, psq[4] = {0.f, 0.f, 0.f, 0.f};

  v8f zz;
  for (int v = 0; v < 8; ++v) zz[v] = 0.f;

  for (int nt = 0; nt < 2560; nt += 64) {
    const int nw = nt + 16 * w;
    if (nw + mr + 64 < 2560)  // prefetch next tile's A rows
      __builtin_prefetch(An + (size_t)(nw + mr + 64) * 300, 0, 0);
    v8f acc[4];
#pragma unroll
    for (int u = 0; u < 4; ++u) acc[u] = zz;

    for (int c0 = 0; c0 < 300; c0 += 4) {
      const int kc = c0 + 2 * half;
      v2f a = *(const v2f*)(An + (size_t)(nw + mr) * 300 + kc);
      a.x *= gl[kc];
      a.y *= gl[kc + 1];
#pragma unroll
      for (int u = 0; u < 4; ++u) {
        v2f bf = *(const v2f*)(Wwp + (size_t)(o0 + 16 * u + mr) * 300 + kc);
        acc[u] = __builtin_amdgcn_wmma_f32_16x16x4_f32(
            false, a, false, bf, (short)0, acc[u], false, false);
      }
    }

    // epilogue: leaky + bias, running sums, per-(b,o) max/min
    const int r0 = nw + 8 * half;      // first of this lane's 8 rows
    const int b0 = r0 / 40;
    const int bend = (r0 + 7) / 40;
    const int splitv = (bend > b0) ? (40 * (b0 + 1) - r0) : 8;
#pragma unroll
    for (int u = 0; u < 4; ++u) {
      float mx0 = -__builtin_inff(), mn0 = __builtin_inff();
      float mx1 = -__builtin_inff(), mn1 = __builtin_inff();
#pragma unroll
      for (int v = 0; v < 8; ++v) {
        float x = acc[u][v] + bp[u];
        x = LEAKY(x);
        psum[u] += x;
        psq[u] += x * x;
        if (v < splitv) { mx0 = fmaxf(mx0, x); mn0 = fminf(mn0, x); }
        else            { mx1 = fmaxf(mx1, x); mn1 = fminf(mn1, x); }
      }
      const int ol = 16 * u + mr;
      atomicMax(&mxk[b0 * 64 + ol], fenc(mx0));
      atomicMin(&mnk[b0 * 64 + ol], fenc(mn0));
      if (splitv < 8) {
        atomicMax(&mxk[(b0 + 1) * 64 + ol], fenc(mx1));
        atomicMin(&mnk[(b0 + 1) * 64 + ol], fenc(mn1));
      }
    }
  }

#pragma unroll
  for (int u = 0; u < 4; ++u) {
    psum[u] += __shfl_xor(psum[u], 16, 32);
    psq[u] += __shfl_xor(psq[u], 16, 32);
    if (lane < 16) {
      atomicAdd(&osum[16 * u + mr], psum[u]);
      atomicAdd(&osq[16 * u + mr], psq[u]);
    }
  }
  __syncthreads();

  if (tid < 64) {  // BN stats + conditional affine coefficients per o
    float m = osum[tid] * (1.0f / 2560.0f);
    float var = osq[tid] * (1.0f / 2560.0f) - m * m;
    float inv = rsqrtf(var + 1e-5f);
    int o = o0 + tid;
    float s = (1.0f + g2arr[(size_t)i * 1024 + o]) * inv;
    float cc = b2arr[(size_t)i * 1024 + o] - m * s;
    osum[tid] = s;
    osq[tid] = cc;
  }
  __syncthreads();

  {  // max over t via sign(s), then partial l2 dot/ssq per caption b
    const int b = tid >> 1;
    const int olbase = (tid & 1) * 32;
    float d = 0.f, q = 0.f;
    for (int k = 0; k < 32; ++k) {
      int ol = olbase + k;
      float s = osum[ol], cc = osq[ol];
      float mxv = fdec(mxk[b * 64 + ol]);
      float mnv = fdec(mnk[b * 64 + ol]);
      float om = (s >= 0.f ? s * mxv : s * mnv) + cc;
      d += iv[(size_t)i * 1024 + o0 + ol] * om;
      q += om * om;
    }
    d += __shfl_xor(d, 1, 32);
    q += __shfl_xor(q, 1, 32);
    if ((tid & 1) == 0) {
      atomicAdd(&dotp[i * 64 + b], d);
      atomicAdd(&ssq[i * 64 + b], q);
    }
  }
}

__global__ void finalize_k(const float* __restrict__ dotp, const float* __restrict__ ssq,
                           float* __restrict__ out) {
  int idx = blockIdx.x * blockDim.x + threadIdx.x;
  if (idx < 64 * 64) out[idx] = dotp[idx] * rsqrtf(ssq[idx]);
}

// ---------------------------------------------------------------------------
extern "C" void kernel_launch(void* const* d_in, const int* in_sizes, int n_in,
                              void* d_out, int out_size, void* d_ws, size_t ws_size,
                              hipStream_t stream) {
  const float* img_embed = (const float*)d_in[0];
  const float* cap_embed = (const float*)d_in[1];
  const float* Wimg = (const float*)d_in[2];
  const float* bimg = (const float*)d_in[3];
  const float* Wq_i = (const float*)d_in[4];
  const float* bq_i = (const float*)d_in[5];
  const float* Wk_i = (const float*)d_in[6];
  const float* bk_i = (const float*)d_in[7];
  const float* Wv_i = (const float*)d_in[8];
  const float* bv_i = (const float*)d_in[9];
  const float* gamma_i = (const float*)d_in[10];
  const float* Wq_w = (const float*)d_in[11];
  const float* bq_w = (const float*)d_in[12];
  const float* Wk_w = (const float*)d_in[13];
  const float* bk_w = (const float*)d_in[14];
  const float* Wv_w = (const float*)d_in[15];
  const float* bv_w = (const float*)d_in[16];
  const float* gamma_w = (const float*)d_in[17];
  const float* Wg1a = (const float*)d_in[18];
  const float* bg1a = (const float*)d_in[19];
  const float* Wg1b = (const float*)d_in[20];
  const float* bg1b = (const float*)d_in[21];
  const float* Wb1a = (const float*)d_in[22];
  const float* bb1a = (const float*)d_in[23];
  const float* Wb1b = (const float*)d_in[24];
  const float* bb1b = (const float*)d_in[25];
  const float* Wwp = (const float*)d_in[26];
  const float* bwp = (const float*)d_in[27];
  const float* Wg2a = (const float*)d_in[28];
  const float* bg2a = (const float*)d_in[29];
  const float* Wg2b = (const float*)d_in[30];
  const float* bg2b = (const float*)d_in[31];
  const float* Wb2a = (const float*)d_in[32];
  const float* bb2a = (const float*)d_in[33];
  const float* Wb2b = (const float*)d_in[34];
  const float* bb2b = (const float*)d_in[35];
  // d_in[36] = lens : unused by the reference.

  float* ws = (float*)d_ws;
  size_t off = 0;
  auto alloc = [&](size_t n) { size_t p = off; off += (n + 15) & ~(size_t)15; return ws + p; };

  float* X = alloc(2304 * 1024);      // leaky(img conv)  (n=b*36+t, o)
  float* Q = alloc(2304 * 128);
  float* Km = alloc(2304 * 128);
  float* V = alloc(2304 * 1024);
  float* vec = alloc(64 * 1024);
  float* iv = alloc(64 * 1024);
  float* capT = alloc(64 * 300 * 40);
  float* qw = alloc(64 * 75 * 40);
  float* kw = alloc(64 * 75 * 40);
  float* vw = alloc(64 * 300 * 40);
  float* cap2 = alloc(64 * 300 * 40);
  float* chm = alloc(300);
  float* chinv = alloc(300);
  float* An = alloc(2560 * 300);      // norm_cap as (n, c) row-major
  float* g1 = alloc(64 * 300);
  float* b1a = alloc(64 * 300);
  float* g2 = alloc(64 * 1024);
  float* b2a = alloc(64 * 1024);
  float* bpr = alloc(64 * 1024);
  float* dtp = alloc(64 * 64);
  float* sq = alloc(64 * 64);
  (void)in_sizes; (void)n_in; (void)out_size; (void)ws_size;

  // ---- image branch ----
  gemm_nt_wmma<<<dim3(16, 18), 128, 0, stream>>>(img_embed, Wimg, bimg, X, 2304, 1024, 2048, 1);
  gemm_nt_wmma<<<dim3(2, 18), 128, 0, stream>>>(X, Wq_i, bq_i, Q, 2304, 128, 1024, 0);
  gemm_nt_wmma<<<dim3(2, 18), 128, 0, stream>>>(X, Wk_i, bk_i, Km, 2304, 128, 1024, 0);
  gemm_nt_wmma<<<dim3(16, 18), 128, 0, stream>>>(X, Wv_i, bv_i, V, 2304, 1024, 1024, 0);
  img_attn_vec<<<64, 128, 0, stream>>>(Q, Km, V, X, gamma_i, vec);
  iv_norm<<<64, 256, 0, stream>>>(vec, iv);

  // ---- caption branch ----
  build_capT<<<3000, 256, 0, stream>>>(cap_embed, capT);
  conv1x1_ct<<<750, 256, 0, stream>>>(capT, Wq_w, bq_w, qw, 64, 300, 40, 75);
  conv1x1_ct<<<750, 256, 0, stream>>>(capT, Wk_w, bk_w, kw, 64, 300, 40, 75);
  conv1x1_ct<<<3000, 256, 0, stream>>>(capT, Wv_w, bv_w, vw, 64, 300, 40, 300);
  word_attn<<<64, 128, 0, stream>>>(qw, kw, vw, capT, gamma_w, cap2);
  bn_stats<<<300, 256, 0, stream>>>(cap2, chm, chinv);
  build_An<<<3000, 256, 0, stream>>>(cap2, chm, chinv, An);

  // ---- conditioning MLPs ----
  mlp2<<<64, 128, 0, stream>>>(vec, Wg1a, bg1a, Wg1b, bg1b, g1, 300);
  mlp2<<<64, 128, 0, stream>>>(vec, Wb1a, bb1a, Wb1b, bb1b, b1a, 300);
  mlp2<<<64, 128, 0, stream>>>(vec, Wg2a, bg2a, Wg2b, bg2b, g2, 1024);
  mlp2<<<64, 128, 0, stream>>>(vec, Wb2a, bb2a, Wb2b, bb2b, b2a, 1024);
  bprime_k<<<dim3(8, 64), 128, 0, stream>>>(Wwp, bwp, b1a, bpr);

  // ---- fused main GEMM + reductions ----
  zero_k<<<32, 256, 0, stream>>>(dtp, 2 * 64 * 64);  // dtp and sq are contiguous
  cbn_gemm_reduce<<<dim3(16, 64), 128, 0, stream>>>(An, Wwp, g1, g2, b2a, bpr, iv, dtp, sq);
  finalize_k<<<16, 256, 0, stream>>>(dtp, sq, (float*)d_out);
}